// CarrierFrequencyOffset_57303453663816
// MI455X (gfx1250) — compile-verified
//
#include <hip/hip_runtime.h>
#include <hip/hip_bf16.h>
#include <stdint.h>

// CFO rotate: out[b,n] = x[b,n] * exp(j*2*pi*n*wd/Fs), B=64, N=2^18.
// Memory-bound (0.25 flop/byte): optimize the data path, not the math path.

typedef float f32x4 __attribute__((ext_vector_type(4)));

#define B_DIM     64
#define N_DIM     262144                 // 2^18 (power of two -> mask for n)
#define TOTAL_E   (B_DIM * N_DIM)        // 16,777,216 elements per plane
#define VEC       4
#define BLOCK     256                    // 8 wave32
#define TILE_E    (BLOCK * VEC)          // 1024 elements / tile (fits one row)
#define NUM_TILES (TOTAL_E / TILE_E)     // 16384
#define NBLOCKS   2048                   // persistent blocks, 8 tiles each
#define INV_FS    (1.0f / 64.0e9f)

__global__ __launch_bounds__(BLOCK) void cfo_rotate_kernel(
    const float* __restrict__ xr,
    const float* __restrict__ xi,
    const float* __restrict__ wdelta,
    float* __restrict__ out)
{
    // Two-stage double buffer, per-lane-private slots (no barriers needed).
    __shared__ float bufR[2][TILE_E];
    __shared__ float bufI[2][TILE_E];

    const int tid = threadIdx.x;

    // Phase step in *revolutions* per sample: v_sin/v_cos consume 2*pi-scaled
    // arguments, so ang_rev = n * (wd / Fs) needs no 2*pi multiply.
    const float scale = wdelta[0] * INV_FS;

    // SGPR base addresses for the saddr form of the async loads.
    const uint64_t gr = (uint64_t)(uintptr_t)xr;
    const uint64_t gi = (uint64_t)(uintptr_t)xi;

    // LDS byte offsets for this lane's 16B slot in each stage.
    // Generic pointers to __shared__ carry the LDS offset in the low 32 bits.
    uint32_t ldsR[2], ldsI[2];
    ldsR[0] = (uint32_t)(uintptr_t)&bufR[0][tid * VEC];
    ldsR[1] = (uint32_t)(uintptr_t)&bufR[1][tid * VEC];
    ldsI[0] = (uint32_t)(uintptr_t)&bufI[0][tid * VEC];
    ldsI[1] = (uint32_t)(uintptr_t)&bufI[1][tid * VEC];

    // Issue one tile's inbound DMA: 2 async b128 loads per lane (real, imag).
    // ASYNCcnt += 2 per call; async loads complete in order.
    auto issue = [&](int tile, int k) {
        const uint32_t goff =
            (uint32_t)((uint32_t)(tile * TILE_E + tid * VEC) * sizeof(float));
        asm volatile("global_load_async_to_lds_b128 %0, %1, %2"
                     :: "v"(ldsR[k]), "v"(goff), "s"(gr) : "memory");
        asm volatile("global_load_async_to_lds_b128 %0, %1, %2"
                     :: "v"(ldsI[k]), "v"(goff), "s"(gi) : "memory");
    };

    int t = (int)blockIdx.x;
    int k = 0;
    if (t < NUM_TILES) issue(t, k);      // prologue: stage tile 0 for this block

    for (; t < NUM_TILES; t += (int)gridDim.x) {
        const int tn = t + (int)gridDim.x;
        if (tn < NUM_TILES) {
            issue(tn, k ^ 1);            // overlap next tile's DMA with compute
            asm volatile("s_wait_asynccnt 0x2" ::: "memory"); // current tile done
        } else {
            asm volatile("s_wait_asynccnt 0x0" ::: "memory");
        }

        // Consume staged data: ds_load_b128 from this lane's private slot.
        const f32x4 vr = *(const f32x4*)&bufR[k][tid * VEC];
        const f32x4 vi = *(const f32x4*)&bufI[k][tid * VEC];

        const uint32_t e0 = (uint32_t)(t * TILE_E + tid * VEC);
        const float n0 = (float)(e0 & (N_DIM - 1)); // n index (row-local)

        f32x4 orv, oiv;
#pragma unroll
        for (int j = 0; j < VEC; ++j) {
            const float rev = (n0 + (float)j) * scale;        // revolutions
            const float c = __builtin_amdgcn_cosf(rev);       // v_cos_f32
            const float s = __builtin_amdgcn_sinf(rev);       // v_sin_f32
            orv[j] = vr[j] * c - vi[j] * s;
            oiv[j] = vr[j] * s + vi[j] * c;
        }

        // Streaming 134MB output, never re-read: non-temporal b128 stores.
        __builtin_nontemporal_store(orv, (f32x4*)(out + e0));
        __builtin_nontemporal_store(oiv, (f32x4*)(out + TOTAL_E + e0));

        k ^= 1;
    }
}

extern "C" void kernel_launch(void* const* d_in, const int* in_sizes, int n_in,
                              void* d_out, int out_size, void* d_ws, size_t ws_size,
                              hipStream_t stream) {
    (void)in_sizes; (void)n_in; (void)out_size; (void)d_ws; (void)ws_size;
    const float* xr = (const float*)d_in[0];
    const float* xi = (const float*)d_in[1];
    const float* wd = (const float*)d_in[2];
    float* out = (float*)d_out;
    cfo_rotate_kernel<<<NBLOCKS, BLOCK, 0, stream>>>(xr, xi, wd, out);
}